// GCN_27891517620413
// MI455X (gfx1250) — compile-verified
//
#include <hip/hip_runtime.h>
#include <hip/hip_bf16.h>

#define NN 50000
#define NE 800000
#define EPSV 1e-5f

typedef float v2f __attribute__((ext_vector_type(2)));
typedef float v8f __attribute__((ext_vector_type(8)));

// ---------------- degree / norm ----------------
__global__ void k_deg_init(float* deg) {
    int i = blockIdx.x * blockDim.x + threadIdx.x;
    if (i < NN) deg[i] = 1.0f;                 // self-loop
}

__global__ void k_deg_count(const int* __restrict__ dst, float* deg) {
    int e = blockIdx.x * blockDim.x + threadIdx.x;
    if (e < NE) atomicAdd(&deg[dst[e]], 1.0f);
}

__global__ void k_deg_finalize(float* dinv) {
    int i = blockIdx.x * blockDim.x + threadIdx.x;
    if (i < NN) dinv[i] = rsqrtf(dinv[i]);
}

// ---------------- fp32 WMMA GEMM: H[N,Fout] = X[N,Fin] @ W[Fout,Fin]^T ----------------
// 2x2 register blocking per wave: two 16-row M tiles x two 16-col N tiles,
// so each K-step is 4 x global_load_b64 feeding 4 x V_WMMA_F32_16X16X4_F32
// (1 load per WMMA instead of 2 -> 8 flop/B out of L1).
__global__ __launch_bounds__(256) void k_gemm_wmma(const float* __restrict__ X,
                                                   const float* __restrict__ W,
                                                   float* __restrict__ H,
                                                   int Fin, int Fout) {
    const int lane = threadIdx.x & 31;
    const int wave = threadIdx.x >> 5;          // 8 waves / block
    const int half = lane >> 4;                 // K sub-select (0: k0..k1, 1: k2..k3)
    const int r    = lane & 15;                 // M (for A) / N (for B) index
    const int base = blockIdx.x * 32;           // 32-row super-tile of X
    const bool hasM1 = (base + 16) < NN;        // last block may only have 16 rows
    const int row0 = base + r;
    const int row1 = hasM1 ? (base + 16 + r) : r;   // safe dummy rows if absent
    const float* x0row = X + (size_t)row0 * Fin + 2 * half;
    const float* x1row = X + (size_t)row1 * Fin + 2 * half;
    const int ntiles = Fout >> 4;

    for (int nt0 = wave; nt0 < ntiles; nt0 += 16) {
        const int  nt1  = nt0 + 8;
        const bool has2 = nt1 < ntiles;         // Fout==256 -> true, Fout==128 -> false
        const float* w0row = W + (size_t)(nt0 * 16 + r) * Fin + 2 * half;
        const float* w1row = W + (size_t)((has2 ? nt1 : nt0) * 16 + r) * Fin + 2 * half;

        v8f acc00 = {0.f,0.f,0.f,0.f,0.f,0.f,0.f,0.f};
        v8f acc01 = acc00, acc10 = acc00, acc11 = acc00;

        if (has2) {
#pragma unroll 2
            for (int k = 0; k < Fin; k += 4) {
                v2f a0 = *(const v2f*)(x0row + k);
                v2f a1 = *(const v2f*)(x1row + k);
                v2f b0 = *(const v2f*)(w0row + k);
                v2f b1 = *(const v2f*)(w1row + k);
                acc00 = __builtin_amdgcn_wmma_f32_16x16x4_f32(false, a0, false, b0, (short)0, acc00, false, false);
                acc10 = __builtin_amdgcn_wmma_f32_16x16x4_f32(false, a1, false, b0, (short)0, acc10, false, false);
                acc01 = __builtin_amdgcn_wmma_f32_16x16x4_f32(false, a0, false, b1, (short)0, acc01, false, false);
                acc11 = __builtin_amdgcn_wmma_f32_16x16x4_f32(false, a1, false, b1, (short)0, acc11, false, false);
            }
        } else {
#pragma unroll 2
            for (int k = 0; k < Fin; k += 4) {
                v2f a0 = *(const v2f*)(x0row + k);
                v2f a1 = *(const v2f*)(x1row + k);
                v2f b0 = *(const v2f*)(w0row + k);
                acc00 = __builtin_amdgcn_wmma_f32_16x16x4_f32(false, a0, false, b0, (short)0, acc00, false, false);
                acc10 = __builtin_amdgcn_wmma_f32_16x16x4_f32(false, a1, false, b0, (short)0, acc10, false, false);
            }
        }

        // D layout: VGPR i -> (M = i + 8*half, N = r), lanes 0-15 / 16-31
        {
            float* h00 = H + (size_t)(base + 8 * half) * Fout + nt0 * 16 + r;
#pragma unroll
            for (int i = 0; i < 8; ++i) h00[(size_t)i * Fout] = acc00[i];
            if (has2) {
                float* h01 = H + (size_t)(base + 8 * half) * Fout + nt1 * 16 + r;
#pragma unroll
                for (int i = 0; i < 8; ++i) h01[(size_t)i * Fout] = acc01[i];
            }
            if (hasM1) {
                float* h10 = H + (size_t)(base + 16 + 8 * half) * Fout + nt0 * 16 + r;
#pragma unroll
                for (int i = 0; i < 8; ++i) h10[(size_t)i * Fout] = acc10[i];
                if (has2) {
                    float* h11 = H + (size_t)(base + 16 + 8 * half) * Fout + nt1 * 16 + r;
#pragma unroll
                    for (int i = 0; i < 8; ++i) h11[(size_t)i * Fout] = acc11[i];
                }
            }
        }
    }
}

// ---------------- out = bias + dinv^2 * h   (self-loop + bias init) ----------------
__global__ void k_init_out(const float* __restrict__ h, const float* __restrict__ dinv,
                           const float* __restrict__ bias, float* __restrict__ out,
                           int F, int total) {
    int idx = blockIdx.x * blockDim.x + threadIdx.x;
    if (idx >= total) return;
    int i = idx / F, f = idx - i * F;
    float dv = dinv[i];
    out[idx] = bias[f] + dv * dv * h[idx];
}

// ---------------- edge scatter: out[dst] += dinv[src]*dinv[dst] * h[src] ----------------
__global__ __launch_bounds__(256) void k_scatter(const int* __restrict__ src,
                                                 const int* __restrict__ dst,
                                                 const float* __restrict__ dinv,
                                                 const float* __restrict__ h,
                                                 float* out, int F) {
    int e = blockIdx.x * 8 + (threadIdx.x >> 5);   // one wave32 per edge
    if (e >= NE) return;
    int lane = threadIdx.x & 31;
    int s = src[e], d = dst[e];
    float norm = dinv[s] * dinv[d];
    const float* hs = h + (size_t)s * F;
    float* od = out + (size_t)d * F;
    for (int f = lane; f < F; f += 32)
        atomicAdd(&od[f], norm * hs[f]);
}

// ---------------- graph-wide LayerNorm stats ----------------
__global__ void k_zero_stats(float* stats) {
    if (threadIdx.x < 2) stats[threadIdx.x] = 0.0f;
}

__global__ __launch_bounds__(256) void k_reduce(const float* __restrict__ x,
                                                float* stats, int total) {
    __shared__ float sh0[256];
    __shared__ float sh1[256];
    float s = 0.f, s2 = 0.f;
    for (int i = blockIdx.x * blockDim.x + threadIdx.x; i < total;
         i += gridDim.x * blockDim.x) {
        float v = x[i];
        s += v; s2 += v * v;
    }
    sh0[threadIdx.x] = s; sh1[threadIdx.x] = s2;
    __syncthreads();
    for (int w = 128; w > 0; w >>= 1) {
        if (threadIdx.x < w) {
            sh0[threadIdx.x] += sh0[threadIdx.x + w];
            sh1[threadIdx.x] += sh1[threadIdx.x + w];
        }
        __syncthreads();
    }
    if (threadIdx.x == 0) {
        atomicAdd(&stats[0], sh0[0]);
        atomicAdd(&stats[1], sh1[0]);
    }
}

// ---------------- (x - mu)/sqrt(var+eps)*g + be, then PReLU ----------------
__global__ void k_norm_prelu(const float* __restrict__ xin, float* __restrict__ xout,
                             const float* __restrict__ g, const float* __restrict__ be,
                             const float* __restrict__ a, const float* __restrict__ stats,
                             int F, int total) {
    int idx = blockIdx.x * blockDim.x + threadIdx.x;
    if (idx >= total) return;
    float inv_n = 1.0f / (float)total;
    float mu  = stats[0] * inv_n;
    float var = stats[1] * inv_n - mu * mu;
    float rstd = rsqrtf(var + EPSV);
    int f = idx - (idx / F) * F;
    float v = (xin[idx] - mu) * rstd * g[f] + be[f];
    xout[idx] = v >= 0.f ? v : a[0] * v;
}

extern "C" void kernel_launch(void* const* d_in, const int* in_sizes, int n_in,
                              void* d_out, int out_size, void* d_ws, size_t ws_size,
                              hipStream_t stream) {
    (void)in_sizes; (void)n_in; (void)out_size; (void)ws_size;
    const float* x  = (const float*)d_in[0];
    const int*   ei = (const int*)d_in[1];
    const int*   src = ei;
    const int*   dst = ei + NE;
    const float* W[3]  = {(const float*)d_in[2],  (const float*)d_in[7],  (const float*)d_in[12]};
    const float* b[3]  = {(const float*)d_in[3],  (const float*)d_in[8],  (const float*)d_in[13]};
    const float* g[3]  = {(const float*)d_in[4],  (const float*)d_in[9],  (const float*)d_in[14]};
    const float* be[3] = {(const float*)d_in[5],  (const float*)d_in[10], (const float*)d_in[15]};
    const float* a[3]  = {(const float*)d_in[6],  (const float*)d_in[11], (const float*)d_in[16]};

    char*  ws    = (char*)d_ws;
    float* dinv  = (float*)ws;                                     // NN floats
    float* stats = (float*)(ws + 204800);                          // 2 floats
    float* bufH  = (float*)(ws + (1 << 20));                       // NN*256 floats
    float* bufA  = (float*)(ws + (1 << 20) + (size_t)NN * 256 * 4);// NN*256 floats

    // dinv = rsqrt(1 + in-degree)
    k_deg_init    <<<(NN + 255) / 256, 256, 0, stream>>>(dinv);
    k_deg_count   <<<(NE + 255) / 256, 256, 0, stream>>>(dst, dinv);
    k_deg_finalize<<<(NN + 255) / 256, 256, 0, stream>>>(dinv);

    const int Fdims[4] = {128, 256, 256, 128};
    const float* cur = x;
    for (int L = 0; L < 3; ++L) {
        int Fin = Fdims[L], Fout = Fdims[L + 1];
        int total = NN * Fout;
        float* out = (L == 2) ? (float*)d_out : bufA;

        k_gemm_wmma <<<(NN + 31) / 32, 256, 0, stream>>>(cur, W[L], bufH, Fin, Fout);
        k_init_out  <<<(total + 255) / 256, 256, 0, stream>>>(bufH, dinv, b[L], out, Fout, total);
        k_scatter   <<<NE / 8, 256, 0, stream>>>(src, dst, dinv, bufH, out, Fout);
        k_zero_stats<<<1, 32, 0, stream>>>(stats);
        k_reduce    <<<1024, 256, 0, stream>>>(out, stats, total);
        k_norm_prelu<<<(total + 255) / 256, 256, 0, stream>>>(out, out, g[L], be[L], a[L],
                                                              stats, Fout, total);
        cur = out;
    }
}